// DRN_GloVe_9878424781100
// MI455X (gfx1250) — compile-verified
//
#include <hip/hip_runtime.h>
#include <hip/hip_bf16.h>
#include <math.h>

// ---------------- constants from the reference ----------------
#define Bsz 16
#define Ssz 512
#define DE 100
#define DT 20
#define DI 20
#define INsz 140
#define Hsz 128
#define GD 256
#define BANK 768
#define Nn 128
#define Ee 48
#define Pp 512
#define Rr 97
#define CLS 1556   // BANK*2 + 20
#define G4H 512    // 4*H

typedef __attribute__((ext_vector_type(2))) float v2f;
typedef __attribute__((ext_vector_type(8))) float v8f;

__device__ __forceinline__ v8f wmma4(v2f a, v2f b, v8f c) {
  // D = A(16x4) * B(4x16) + C, f32. 8-arg form: (neg_a, A, neg_b, B, c_mod, C, reuse_a, reuse_b)
  return __builtin_amdgcn_wmma_f32_16x16x4_f32(false, a, false, b, (short)0, c, false, false);
}

__device__ __forceinline__ float sigf(float x) { return 1.0f / (1.0f + expf(-x)); }

// ---------------- generic tiled WMMA GEMM ----------------
// C[bz] = act( A[bz](MxK) * B[bz](KxN) + bias ), row-major with leading dims.
// block: 256 threads (8 waves, 2x4 wave grid), tile 64x64, K staged by 16.
// Tile loads take a uniform fast path (no per-lane guards) when the whole
// tile is in-bounds; ragged edges (K remainder, N=97, M=48) use guarded path.
#define TILE 64
#define KT 16

__global__ __launch_bounds__(256) void gemm_wmma_k(
    const float* __restrict__ A, const float* __restrict__ Bm,
    const float* __restrict__ bias, float* __restrict__ C,
    int M, int N, int K, int lda, int ldb, int ldc,
    long sA, long sB, long sC, int act)
{
  __shared__ float As[TILE][KT + 2];   // As[m][k]
  __shared__ float Bs[TILE][KT + 2];   // Bs[n][k]  (B tile stored transposed)
  const int bz = blockIdx.z;
  A  += (long)bz * sA;
  Bm += (long)bz * sB;
  C  += (long)bz * sC;
  const int m0 = blockIdx.y * TILE;
  const int n0 = blockIdx.x * TILE;
  const int tid = threadIdx.x;
  const int lane = tid & 31;
  const int w  = tid >> 5;
  const int wy = w >> 2;          // 0..1 -> 32-row strip
  const int wx = w & 3;           // 0..3 -> 16-col strip
  const int lh = lane >> 4;       // half-wave select
  const int ll = lane & 15;

  const bool fullM = (m0 + TILE) <= M;
  const bool fullN = (n0 + TILE) <= N;

  // per-thread load coordinates (uniform mapping)
  const int am = (tid * 4) >> 4;      // A: row in tile, 4 consecutive k per thread
  const int ak = (tid * 4) & 15;

  v8f acc0 = {}, acc1 = {};

  for (int k0 = 0; k0 < K; k0 += KT) {
    const bool fullK = (k0 + KT) <= K;

    if (fullM & fullK) {               // uniform fast path: unguarded coalesced loads
      const float* pa = A + (long)(m0 + am) * lda + (k0 + ak);
#pragma unroll
      for (int r = 0; r < 4; ++r) As[am][ak + r] = pa[r];
    } else {
#pragma unroll
      for (int r = 0; r < 4; ++r) {
        int m = am, k = ak + r;
        float v = 0.f;
        if ((m0 + m) < M && (k0 + k) < K) v = A[(long)(m0 + m) * lda + (k0 + k)];
        As[m][k] = v;
      }
    }

    if (fullN & fullK) {
#pragma unroll
      for (int r = 0; r < 4; ++r) {    // B tile 16x64 -> Bs[n][k]
        int idx = tid + 256 * r;
        int k = idx >> 6, n = idx & 63;
        Bs[n][k] = Bm[(long)(k0 + k) * ldb + (n0 + n)];
      }
    } else {
#pragma unroll
      for (int r = 0; r < 4; ++r) {
        int idx = tid + 256 * r;
        int k = idx >> 6, n = idx & 63;
        float v = 0.f;
        if ((k0 + k) < K && (n0 + n) < N) v = Bm[(long)(k0 + k) * ldb + (n0 + n)];
        Bs[n][k] = v;
      }
    }
    __syncthreads();
#pragma unroll
    for (int kk = 0; kk < 4; ++kk) {
      int kb = kk * 4 + lh * 2;
      v2f a0 = { As[wy * 32 + ll][kb],      As[wy * 32 + ll][kb + 1] };
      v2f a1 = { As[wy * 32 + 16 + ll][kb], As[wy * 32 + 16 + ll][kb + 1] };
      v2f b  = { Bs[wx * 16 + ll][kb],      Bs[wx * 16 + ll][kb + 1] };
      acc0 = wmma4(a0, b, acc0);
      acc1 = wmma4(a1, b, acc1);
    }
    __syncthreads();
  }

  const int n = n0 + wx * 16 + ll;
  if (n < N) {
    float bv = bias ? bias[n] : 0.f;
#pragma unroll
    for (int j = 0; j < 8; ++j) {
      int mrow = j + 8 * lh;
      int ma = m0 + wy * 32 + mrow;
      int mb = ma + 16;
      if (ma < M) {
        float v = acc0[j] + bv;
        if (act == 1) v = fmaxf(v, 0.f); else if (act == 2) v = tanhf(v);
        C[(long)ma * ldc + n] = v;
      }
      if (mb < M) {
        float v = acc1[j] + bv;
        if (act == 1) v = fmaxf(v, 0.f); else if (act == 2) v = tanhf(v);
        C[(long)mb * ldc + n] = v;
      }
    }
  }
}

// ---------------- LSTM recurrent scan (one block per direction) ----------------
// gx: [B*S, 512] precomputed x@Wih^T + b. Whh: [512,128]. enc: [B,S,256].
// 512 threads = 16 waves; wave w owns gate columns [32w, 32w+32).
// Whh^T B-fragments held persistently in VGPRs (128 VGPRs/lane).
__global__ __launch_bounds__(512) void lstm_scan_k(
    const float* __restrict__ gx, const float* __restrict__ Whh,
    const int* __restrict__ mask, float* __restrict__ enc,
    int reverse, int outOff)
{
  __shared__ float h_s[Bsz * Hsz];
  __shared__ float c_s[Bsz * Hsz];
  __shared__ float g_s[Bsz * G4H];
  const int tid = threadIdx.x;
  const int lane = tid & 31;
  const int w = tid >> 5;
  const int lh = lane >> 4;
  const int ll = lane & 15;
  const int nb = w * 32;

  for (int i = tid; i < Bsz * Hsz; i += 512) { h_s[i] = 0.f; c_s[i] = 0.f; }

  // persistent B fragments of Whh^T: B[k][n] = Whh[n][k]
  v2f bf[2][32];
#pragma unroll
  for (int t = 0; t < 2; ++t) {
    int n = nb + t * 16 + ll;
#pragma unroll
    for (int kk = 0; kk < 32; ++kk) {
      int kb = kk * 4 + lh * 2;
      bf[t][kk] = *(const v2f*)(Whh + (long)n * Hsz + kb);
    }
  }
  __syncthreads();

  for (int ss = 0; ss < Ssz; ++ss) {
    int t = reverse ? (Ssz - 1 - ss) : ss;
    v8f acc[2];
#pragma unroll
    for (int i = 0; i < 2; ++i) {
      int n = nb + i * 16 + ll;
#pragma unroll
      for (int j = 0; j < 8; ++j) {
        int m = j + 8 * lh;                        // batch row
        acc[i][j] = gx[((long)(m * Ssz + t)) * G4H + n];
      }
    }
#pragma unroll
    for (int kk = 0; kk < 32; ++kk) {
      int kb = kk * 4 + lh * 2;
      v2f a = *(const v2f*)(&h_s[ll * Hsz + kb]);  // A[m=ll][kb..kb+1]
      acc[0] = wmma4(a, bf[0][kk], acc[0]);
      acc[1] = wmma4(a, bf[1][kk], acc[1]);
    }
#pragma unroll
    for (int i = 0; i < 2; ++i) {
      int n = nb + i * 16 + ll;
#pragma unroll
      for (int j = 0; j < 8; ++j) {
        int m = j + 8 * lh;
        g_s[m * G4H + n] = acc[i][j];
      }
    }
    __syncthreads();
    // elementwise LSTM cell update: 16*128 = 2048 states, 4 per thread
#pragma unroll
    for (int r = 0; r < 4; ++r) {
      int p = tid + 512 * r;
      int m = p >> 7, j = p & 127;
      float ig = sigf(g_s[m * G4H + j]);
      float fg = sigf(g_s[m * G4H + 128 + j]);
      float gg = tanhf(g_s[m * G4H + 256 + j]);
      float og = sigf(g_s[m * G4H + 384 + j]);
      float c = fg * c_s[m * Hsz + j] + ig * gg;
      c_s[m * Hsz + j] = c;
      float h = og * tanhf(c);
      h_s[m * Hsz + j] = h;
      enc[((long)(m * Ssz + t)) * GD + outOff + j] = mask[m * Ssz + t] ? h : 0.f;
    }
    __syncthreads();
  }
}

// ---------------- elementwise / gather kernels ----------------
__global__ void embed_k(const int* __restrict__ words, const int* __restrict__ etype,
                        const int* __restrict__ eid, const float* __restrict__ wt,
                        const float* __restrict__ tt, const float* __restrict__ it,
                        float* __restrict__ src) {
  int i = blockIdx.x * blockDim.x + threadIdx.x;
  if (i >= Bsz * Ssz * INsz) return;
  int d = i % INsz, bs = i / INsz;
  float v;
  if (d < DE)            v = wt[(long)words[bs] * DE + d];
  else if (d < DE + DT)  v = tt[etype[bs] * DT + (d - DE)];
  else                   v = it[eid[bs] * DI + (d - DE - DT)];
  src[i] = v;
}

__global__ void transpose_k(const float* __restrict__ in, float* __restrict__ out,
                            int rows, int cols) {
  int i = blockIdx.x * blockDim.x + threadIdx.x;
  if (i >= rows * cols) return;
  int r = i / cols, c = i % cols;
  out[c * rows + r] = in[i];
}

__global__ void span_k(const int* __restrict__ ginfo, const int* __restrict__ gnum,
                       const float* __restrict__ enc, float* __restrict__ feat) {
  int i = blockIdx.x * blockDim.x + threadIdx.x;
  if (i >= Bsz * Nn * GD) return;
  int d = i & (GD - 1);
  int n = (i >> 8) & (Nn - 1);
  int b = i >> 15;
  const int* gi = ginfo + (b * Nn + n) * 4;
  int st = gi[0], en = gi[1];
  float s = 0.f;
  for (int x = st; x < en; ++x) s += enc[((long)(b * Ssz + x)) * GD + d];
  int len = en - st; if (len < 1) len = 1;
  float v = (n < gnum[b]) ? (s / (float)len) : 0.f;
  feat[((long)(b * Nn + n)) * BANK + d] = v;   // slice 0 of feat
}

__global__ void anorm_k(const int* __restrict__ adj, const int* __restrict__ gnum,
                        float* __restrict__ Anorm) {
  int bm = blockIdx.x * blockDim.x + threadIdx.x;
  if (bm >= Bsz * Nn) return;
  int b = bm >> 7, m = bm & (Nn - 1);
  int num = gnum[b];
  bool mok = m < num;
  const int* row = adj + (long)(b * Nn + m) * Nn;
  float cnt = 0.f;
  for (int n = 0; n < Nn; ++n) cnt += (mok && row[n] > 0 && n < num) ? 1.f : 0.f;
  float inv = 1.f / fmaxf(cnt, 1.f);
  float* out = Anorm + (long)(b * Nn + m) * Nn;
  for (int n = 0; n < Nn; ++n) out[n] = (mok && row[n] > 0 && n < num) ? inv : 0.f;
}

__global__ void sel_k(const int* __restrict__ ginfo, const int* __restrict__ gnum,
                      float* __restrict__ sel) {
  int be = blockIdx.x * blockDim.x + threadIdx.x;
  if (be >= Bsz * Ee) return;
  int b = be / Ee, e = be % Ee;
  int num = gnum[b];
  float cnt = 0.f;
  for (int n = 0; n < Nn; ++n) {
    const int* gi = ginfo + (b * Nn + n) * 4;
    if (gi[3] == 2 && n < num && gi[2] == e) cnt += 1.f;
  }
  float inv = 1.f / fmaxf(cnt, 1.f);
  float* out = sel + (long)(b * Ee + e) * Nn;
  for (int n = 0; n < Nn; ++n) {
    const int* gi = ginfo + (b * Nn + n) * 4;
    out[n] = (gi[3] == 2 && n < num && gi[2] == e) ? inv : 0.f;
  }
}

__global__ void rel_k(const float* __restrict__ ebank, const int* __restrict__ pairs,
                      const int* __restrict__ dist, const float* __restrict__ dtab,
                      float* __restrict__ rel) {
  long i = (long)blockIdx.x * blockDim.x + threadIdx.x;
  if (i >= (long)Bsz * Pp * CLS) return;
  int d = (int)(i % CLS);
  long bp = i / CLS;
  int b = (int)(bp / Pp);
  float v;
  if (d < BANK) {
    int h = pairs[bp * 2 + 0];
    v = ebank[((long)(b * Ee + h)) * BANK + d];
  } else if (d < 2 * BANK) {
    int t = pairs[bp * 2 + 1];
    v = ebank[((long)(b * Ee + t)) * BANK + (d - BANK)];
  } else {
    v = dtab[dist[bp] * 20 + (d - 2 * BANK)];
  }
  rel[i] = v;
}

// ---------------- host-side orchestration ----------------
static inline void launch_gemm(hipStream_t s, const float* A, const float* Bm,
                               const float* bias, float* C, int M, int N, int K,
                               int lda, int ldb, int ldc,
                               long sA, long sB, long sC, int batch, int act) {
  dim3 g((unsigned)((N + TILE - 1) / TILE), (unsigned)((M + TILE - 1) / TILE), (unsigned)batch);
  gemm_wmma_k<<<g, 256, 0, s>>>(A, Bm, bias, C, M, N, K, lda, ldb, ldc, sA, sB, sC, act);
}

extern "C" void kernel_launch(void* const* d_in, const int* in_sizes, int n_in,
                              void* d_out, int out_size, void* d_ws, size_t ws_size,
                              hipStream_t stream) {
  (void)in_sizes; (void)n_in; (void)out_size; (void)ws_size;
  const int* words  = (const int*)d_in[0];
  const int* mask   = (const int*)d_in[1];
  const int* etype  = (const int*)d_in[2];
  const int* eid    = (const int*)d_in[3];
  // d_in[4] src_lengths unused (mask already encodes it)
  const int* adj    = (const int*)d_in[5];
  const int* ginfo  = (const int*)d_in[6];
  const int* gnum   = (const int*)d_in[7];
  const int* pairs  = (const int*)d_in[8];
  const int* dist   = (const int*)d_in[9];
  const float* wt   = (const float*)d_in[10];
  const float* tt   = (const float*)d_in[11];
  const float* it   = (const float*)d_in[12];
  const float* Wih_f = (const float*)d_in[13];
  const float* Whh_f = (const float*)d_in[14];
  const float* b_f   = (const float*)d_in[15];
  const float* Wih_b = (const float*)d_in[16];
  const float* Whh_b = (const float*)d_in[17];
  const float* b_b   = (const float*)d_in[18];
  const float* gW1   = (const float*)d_in[19];
  const float* gb1   = (const float*)d_in[20];
  const float* gW2   = (const float*)d_in[21];
  const float* gb2   = (const float*)d_in[22];
  const float* dtab  = (const float*)d_in[23];
  const float* pW1   = (const float*)d_in[24];
  const float* pb1   = (const float*)d_in[25];
  const float* pW2   = (const float*)d_in[26];
  const float* pb2   = (const float*)d_in[27];
  float* out = (float*)d_out;

  float* ws = (float*)d_ws;
  float* src   = ws + 0;          //  1,146,880  [B*S,140]
  float* WihTf = ws + 1200000;    //     71,680  [140,512]
  float* WihTb = ws + 1280000;    //     71,680
  float* gxf   = ws + 1400000;    //  4,194,304  [B*S,512]
  float* gxb   = ws + 5600000;    //  4,194,304
  float* enc   = ws + 9800000;    //  2,097,152  [B,S,256]
  float* feat  = ws + 11900000;   //  1,572,864  [B,N,768]
  float* Anorm = ws + 13500000;   //    262,144  [B,N,N]
  float* tmp   = ws + 13800000;   //    524,288  [B,N,256]
  float* sel   = ws + 14350000;   //     98,304  [B,E,N]
  float* ebank = ws + 14450000;   //    589,824  [B,E,768]
  float* rel   = ws + 15050000;   // 12,746,752  [B*P,1556]
  float* hidden = ws + 0;         // 12,582,912  reuses dead src/gx/enc region

  const int M = Bsz * Ssz;  // 8192

  // 1) embeddings -> src
  embed_k<<<(Bsz * Ssz * INsz + 255) / 256, 256, 0, stream>>>(words, etype, eid, wt, tt, it, src);

  // 2) transpose Wih for both directions
  transpose_k<<<(G4H * INsz + 255) / 256, 256, 0, stream>>>(Wih_f, WihTf, G4H, INsz);
  transpose_k<<<(G4H * INsz + 255) / 256, 256, 0, stream>>>(Wih_b, WihTb, G4H, INsz);

  // 3) gates_x = src @ Wih^T + b  (WMMA)
  launch_gemm(stream, src, WihTf, b_f, gxf, M, G4H, INsz, INsz, G4H, G4H, 0, 0, 0, 1, 0);
  launch_gemm(stream, src, WihTb, b_b, gxb, M, G4H, INsz, INsz, G4H, G4H, 0, 0, 0, 1, 0);

  // 4) recurrent scans (fwd then bwd, one block each, serialized on one stream)
  lstm_scan_k<<<1, 512, 0, stream>>>(gxf, Whh_f, mask, enc, 0, 0);
  lstm_scan_k<<<1, 512, 0, stream>>>(gxb, Whh_b, mask, enc, 1, Hsz);

  // 5) span means -> feat slice 0
  span_k<<<(Bsz * Nn * GD + 255) / 256, 256, 0, stream>>>(ginfo, gnum, enc, feat);

  // 6) normalized adjacency
  anorm_k<<<(Bsz * Nn + 255) / 256, 256, 0, stream>>>(adj, gnum, Anorm);

  // 7) GCN iter 1: tmp = Anorm @ feat[:,:,0:256]; h1 = relu(tmp @ W1 + b1) -> feat slice 1
  launch_gemm(stream, Anorm, feat, nullptr, tmp, Nn, GD, Nn, Nn, BANK, GD,
              (long)Nn * Nn, (long)Nn * BANK, (long)Nn * GD, Bsz, 0);
  launch_gemm(stream, tmp, gW1, gb1, feat + GD, Bsz * Nn, GD, GD, GD, GD, BANK,
              0, 0, 0, 1, 1);

  // 8) GCN iter 2: tmp = Anorm @ feat[:,:,256:512]; h2 = relu(tmp @ W2 + b2) -> feat slice 2
  launch_gemm(stream, Anorm, feat + GD, nullptr, tmp, Nn, GD, Nn, Nn, BANK, GD,
              (long)Nn * Nn, (long)Nn * BANK, (long)Nn * GD, Bsz, 0);
  launch_gemm(stream, tmp, gW2, gb2, feat + 2 * GD, Bsz * Nn, GD, GD, GD, GD, BANK,
              0, 0, 0, 1, 1);

  // 9) entity bank = sel @ feat  (batched)
  sel_k<<<(Bsz * Ee + 255) / 256, 256, 0, stream>>>(ginfo, gnum, sel);
  launch_gemm(stream, sel, feat, nullptr, ebank, Ee, BANK, Nn, Nn, BANK, BANK,
              (long)Ee * Nn, (long)Nn * BANK, (long)Ee * BANK, Bsz, 0);

  // 10) rel = [h_ent | t_ent | dis_embed]
  {
    long total = (long)Bsz * Pp * CLS;
    rel_k<<<(unsigned)((total + 255) / 256), 256, 0, stream>>>(ebank, pairs, dist, dtab, rel);
  }

  // 11) predict MLP: hidden = tanh(rel @ pW1 + pb1); out = hidden @ pW2 + pb2
  launch_gemm(stream, rel, pW1, pb1, hidden, M, 2 * BANK, CLS, CLS, 2 * BANK, 2 * BANK,
              0, 0, 0, 1, 2);
  launch_gemm(stream, hidden, pW2, pb2, out, M, Rr, 2 * BANK, 2 * BANK, Rr, Rr,
              0, 0, 0, 1, 0);
}